// StructureLearner_80041010528662
// MI455X (gfx1250) — compile-verified
//
#include <hip/hip_runtime.h>

// ---------------- problem constants ----------------
#define NN   10000      // nodes
#define DD   512        // feature dim
#define KP1  31         // k+1 (top list size, includes self)
#define CTILES 625      // NN/16 column tiles
#define PASSES 79       // ceil(625 / 8 waves)

typedef __attribute__((ext_vector_type(2))) float v2f;
typedef __attribute__((ext_vector_type(8))) float v8f;

// ---------------- workspace layout (bytes) ----------------
#define OFF_XN        0ull                          // NN*DD f32 = 20,480,000
#define OFF_VALS      20480000ull                   // NN*31 f32 = 1,240,000
#define OFF_INDS      21720000ull                   // NN*31 i32 = 1,240,000
#define OFF_NROW      22960000ull                   // NN f32
#define OFF_NCOL      23000000ull                   // NN f32   (contiguous with deg)
#define OFF_DEG       23040000ull                   // NN f32
#define OFF_DINV      23080000ull                   // NN f32
#define OFF_EVAL      23120000ull                   // NN*31 f32
// total: 24,360,000 bytes

// ---------------- zero fill ----------------
__global__ void zero_f32(float4* __restrict__ p, long long n4) {
  long long i = (long long)blockIdx.x * blockDim.x + threadIdx.x;
  long long stride = (long long)gridDim.x * blockDim.x;
  float4 z; z.x = 0.f; z.y = 0.f; z.z = 0.f; z.w = 0.f;
  for (; i < n4; i += stride) p[i] = z;
}

// ---------------- row normalization ----------------
__global__ __launch_bounds__(256) void normalize_rows(const float* __restrict__ x,
                                                      float* __restrict__ xn) {
  __shared__ float red[8];
  const int row = blockIdx.x;
  const int tid = threadIdx.x;
  const float* xr = x + (size_t)row * DD;
  float a0 = xr[tid];
  float a1 = xr[tid + 256];
  float s = a0 * a0 + a1 * a1;
  #pragma unroll
  for (int o = 16; o > 0; o >>= 1) s += __shfl_down(s, o, 32);
  if ((tid & 31) == 0) red[tid >> 5] = s;
  __syncthreads();
  if (tid == 0) {
    float t = 0.f;
    #pragma unroll
    for (int i = 0; i < 8; ++i) t += red[i];
    red[0] = 1.0f / sqrtf(t);
  }
  __syncthreads();
  const float rinv = red[0];
  float* xo = xn + (size_t)row * DD;
  xo[tid]       = a0 * rinv;
  xo[tid + 256] = a1 * rinv;
}

// ---------------- fused WMMA GEMM + top-(k+1) ----------------
// One block = 16 rows. 8 waves; each wave computes one 16x16 sims tile per
// pass with V_WMMA_F32_16X16X4_F32, then ballot-based top-31 maintenance.
__global__ __launch_bounds__(256) void sims_topk(const float* __restrict__ xn,
                                                 float* __restrict__ vals,
                                                 int*   __restrict__ inds,
                                                 float* __restrict__ norm_row,
                                                 float* __restrict__ norm_col) {
  __shared__ float Atile[16][DD + 4];     // +4 pad -> conflict-free stride-512 access
  __shared__ float simsLds[16][132];      // 16 rows x 128 cols staging (+4 pad)
  __shared__ float topv[16][32];
  __shared__ int   topi[16][32];

  const int tid  = threadIdx.x;
  const int lane = tid & 31;
  const int wid  = tid >> 5;
  const int rowbase = blockIdx.x * 16;

  // load 16x512 A tile (row block of xn), coalesced
  for (int idx = tid; idx < 16 * DD; idx += 256) {
    int r  = idx >> 9;          // / 512
    int kk = idx & (DD - 1);
    Atile[r][kk] = xn[(size_t)(rowbase + r) * DD + kk];
  }
  // init top lists
  for (int idx = tid; idx < 16 * 32; idx += 256) {
    topv[idx >> 5][idx & 31] = -__builtin_inff();
    topi[idx >> 5][idx & 31] = 0;
  }
  __syncthreads();

  // WMMA fp32 16x16x4 per-lane A/B fragment addressing:
  // lanes 0-15: rows/cols m=lane, K = kk+0,kk+1 ; lanes 16-31: m=lane-16, K = kk+2,kk+3
  const int m    = lane & 15;
  const int kofs = (lane >> 4) << 1;   // 0 or 2
  const int hi   = lane >> 4;

  for (int p = 0; p < PASSES; ++p) {
    const int ct = p * 8 + wid;        // wave-uniform
    if (ct < CTILES) {
      const int cb = ct * 16;
      const float* bp = xn + (size_t)(cb + m) * DD + kofs;  // B[k][n] = xn[cb+n][k]
      const float* ap = &Atile[m][kofs];
      v8f c = {0.f, 0.f, 0.f, 0.f, 0.f, 0.f, 0.f, 0.f};
      #pragma unroll 4
      for (int kk = 0; kk < DD; kk += 4) {
        v2f a = *(const v2f*)(ap + kk);   // ds_load_b64
        v2f b = *(const v2f*)(bp + kk);   // global_load_b64 (L2-resident)
        c = __builtin_amdgcn_wmma_f32_16x16x4_f32(
                false, a, false, b, (short)0, c, false, false);
      }
      // C/D layout: VGPR r = row (r + 8*hi), col = lane&15
      #pragma unroll
      for (int r = 0; r < 8; ++r)
        simsLds[r + 8 * hi][wid * 16 + m] = c[r];
    }
    __syncthreads();

    // ---- top-31 maintenance over the fresh 16x(<=128) sims slab ----
    const int colsBase  = p * 128;
    int colsValid = NN - colsBase;
    if (colsValid > 128) colsValid = 128;

    for (int rr = 0; rr < 2; ++rr) {
      const int row = wid + rr * 8;       // 16 rows over 8 waves
      for (int j0 = 0; j0 < 128; j0 += 32) {
        const int j = j0 + lane;
        const float v = (j < colsValid) ? simsLds[row][j] : -__builtin_inff();
        unsigned long long mask = __ballot(v > topv[row][KP1 - 1]);
        while (mask) {
          const int s = __ffsll(mask) - 1;
          mask &= mask - 1;
          const float vv = __shfl(v, s, 32);
          if (vv > topv[row][KP1 - 1]) {     // recheck: threshold may have risen
            if (lane == 0) {                 // serial sorted insertion (rare)
              int pos = KP1 - 1;
              while (pos > 0 && topv[row][pos - 1] < vv) {
                topv[row][pos] = topv[row][pos - 1];
                topi[row][pos] = topi[row][pos - 1];
                --pos;
              }
              topv[row][pos] = vv;
              topi[row][pos] = colsBase + j0 + s;
            }
          }
        }
      }
    }
    __syncthreads();
  }

  // ---- emit vals / inds / norm_row, accumulate norm_col ----
  for (int idx = tid; idx < 16 * KP1; idx += 256) {
    const int r = idx / KP1;
    const int j = idx % KP1;
    const int gr = rowbase + r;
    const float v = topv[r][j];
    const int  ci = topi[r][j];
    vals[(size_t)gr * KP1 + j] = v;
    inds[(size_t)gr * KP1 + j] = ci;
    atomicAdd(&norm_col[ci], v);
  }
  __syncthreads();
  if (tid < 16) {
    float s = 0.f;
    #pragma unroll
    for (int j = 0; j < KP1; ++j) s += topv[tid][j];
    norm_row[rowbase + tid] = s;
  }
}

// ---------------- edge renormalization + degree accumulation ----------------
__global__ void finalize_edges(const float* __restrict__ vals,
                               const int*   __restrict__ inds,
                               const float* __restrict__ norm_row,
                               const float* __restrict__ norm_col,
                               float* __restrict__ edge_val,
                               float* __restrict__ deg) {
  const int e = blockIdx.x * blockDim.x + threadIdx.x;
  if (e >= NN * KP1) return;
  const int r = e / KP1;
  const int c = inds[e];
  const float v  = vals[e];
  const float nr = norm_row[r] + norm_col[r];
  const float nc = norm_row[c] + norm_col[c];
  float val = v * rsqrtf(nr) * rsqrtf(nc);   // norm**-0.5 ; rsqrt(neg)=NaN like jnp
  if (val != val) val = 0.0f;                // NaN -> 0
  const float rv = fmaxf(val, 0.0f);         // relu (symmetrized value)
  edge_val[e] = rv;
  atomicAdd(&deg[r], rv);                    // rows_ = [rows, cols]
  atomicAdd(&deg[c], rv);
}

__global__ void compute_dinv(const float* __restrict__ deg, float* __restrict__ dinv) {
  const int i = blockIdx.x * blockDim.x + threadIdx.x;
  if (i >= NN) return;
  const float d = deg[i];
  dinv[i] = (d > 0.0f) ? rsqrtf(d) : 0.0f;   // deg**-0.5, inf -> 0
}

// ---------------- densified symmetric scatter ----------------
__global__ void scatter_out(const float* __restrict__ edge_val,
                            const int*   __restrict__ inds,
                            const float* __restrict__ dinv,
                            float* __restrict__ out) {
  const int e = blockIdx.x * blockDim.x + threadIdx.x;
  if (e >= NN * KP1) return;
  const int r = e / KP1;
  const int c = inds[e];
  const float nv = edge_val[e] * dinv[r] * dinv[c];
  atomicAdd(&out[(size_t)r * NN + c], nv);
  atomicAdd(&out[(size_t)c * NN + r], nv);
}

// ---------------- host side ----------------
extern "C" void kernel_launch(void* const* d_in, const int* in_sizes, int n_in,
                              void* d_out, int out_size, void* d_ws, size_t ws_size,
                              hipStream_t stream) {
  const float* x = (const float*)d_in[0];  // [NN, DD] f32 ; d_in[1] is k=30 (fixed)
  float* out = (float*)d_out;              // [NN, NN] f32
  char* ws = (char*)d_ws;

  float* xn       = (float*)(ws + OFF_XN);
  float* vals     = (float*)(ws + OFF_VALS);
  int*   inds     = (int*)  (ws + OFF_INDS);
  float* norm_row = (float*)(ws + OFF_NROW);
  float* norm_col = (float*)(ws + OFF_NCOL);
  float* deg      = (float*)(ws + OFF_DEG);
  float* dinv     = (float*)(ws + OFF_DINV);
  float* edge_val = (float*)(ws + OFF_EVAL);

  // zero output (400 MB) and norm_col+deg (contiguous 80 KB)
  zero_f32<<<2048, 256, 0, stream>>>((float4*)out, (long long)NN * NN / 4);
  zero_f32<<<40, 256, 0, stream>>>((float4*)norm_col, (2 * NN) / 4);

  normalize_rows<<<NN, 256, 0, stream>>>(x, xn);

  sims_topk<<<CTILES, 256, 0, stream>>>(xn, vals, inds, norm_row, norm_col);

  const int NE = NN * KP1;
  finalize_edges<<<(NE + 255) / 256, 256, 0, stream>>>(vals, inds, norm_row, norm_col,
                                                       edge_val, deg);
  compute_dinv<<<(NN + 255) / 256, 256, 0, stream>>>(deg, dinv);
  scatter_out<<<(NE + 255) / 256, 256, 0, stream>>>(edge_val, inds, dinv, out);
}